// InteractionBlock_47115791237972
// MI455X (gfx1250) — compile-verified
//
#include <hip/hip_runtime.h>
#include <hip/hip_bf16.h>

#define HD 128
#define NBIL 8
#define LDA 132   // padded LDS row stride (floats) to avoid bank conflicts

typedef float v2f __attribute__((ext_vector_type(2)));
typedef float v8f __attribute__((ext_vector_type(8)));

__device__ __forceinline__ v8f wmma_f32(v2f a, v2f b, v8f c) {
    // D = A(16x4) x B(4x16) + C, fp32 in/out  -> v_wmma_f32_16x16x4_f32
    return __builtin_amdgcn_wmma_f32_16x16x4_f32(
        /*neg_a=*/false, a, /*neg_b=*/false, b,
        /*c_mod=*/(short)0, c, /*reuse_a=*/false, /*reuse_b=*/false);
}

__device__ __forceinline__ float silu_f(float v) {
    return v / (1.0f + __expf(-v));
}

// ---------------------------------------------------------------------------
// Generic [rows x 128] @ [128 x 128] GEMM, A and W staged in LDS.
// Block = 256 threads (8 waves). Block tile = 128 rows x 128 cols.
// Wave w handles rows [16w, 16w+16), loops over the 8 column sub-tiles.
// mode: 0 = none, 1 = silu, 2 = silu then *extra, 3 = silu then +extra
// ---------------------------------------------------------------------------
__global__ __launch_bounds__(256) void gemm_edge(
    const float* __restrict__ A, const float* __restrict__ Aadd,
    const float* __restrict__ W, const float* __restrict__ bias,
    const float* __restrict__ extra, float* __restrict__ out,
    int rows, int mode)
{
    __shared__ float As[128 * LDA];
    __shared__ float Ws[HD * LDA];
    const int tid  = threadIdx.x;
    const int lane = tid & 31;
    const int wv   = tid >> 5;
    const int row0 = blockIdx.x * 128;

    // stage A (zero-padded past `rows`) and W into LDS, float4 at a time
    for (int idx = tid; idx < 128 * 32; idx += 256) {
        int r = idx >> 5; int c = (idx & 31) << 2;
        int gr = row0 + r;
        float4 v = make_float4(0.f, 0.f, 0.f, 0.f);
        if (gr < rows) {
            v = *(const float4*)(A + (size_t)gr * HD + c);
            if (Aadd) {
                float4 u = *(const float4*)(Aadd + (size_t)gr * HD + c);
                v.x += u.x; v.y += u.y; v.z += u.z; v.w += u.w;
            }
        }
        float* p = &As[r * LDA + c];
        p[0] = v.x; p[1] = v.y; p[2] = v.z; p[3] = v.w;
    }
    for (int idx = tid; idx < 128 * 32; idx += 256) {
        int r = idx >> 5; int c = (idx & 31) << 2;
        float4 v = *(const float4*)(W + r * HD + c);
        float* p = &Ws[r * LDA + c];
        p[0] = v.x; p[1] = v.y; p[2] = v.z; p[3] = v.w;
    }
    __syncthreads();

    const int mrow = (wv << 4) + (lane & 15);
    const int koff = (lane < 16) ? 0 : 2;
    const int hsel = (lane < 16) ? 0 : 8;
    const float* arow = &As[mrow * LDA];

    for (int nt = 0; nt < 8; ++nt) {
        const int ncol = (nt << 4) + (lane & 15);
        v8f acc = {};
        #pragma unroll
        for (int k = 0; k < HD; k += 4) {
            v2f a = *(const v2f*)(arow + k + koff);
            v2f b;
            b.x = Ws[(k + koff) * LDA + ncol];
            b.y = Ws[(k + koff + 1) * LDA + ncol];
            acc = wmma_f32(a, b, acc);
        }
        const float bn = bias[ncol];
        #pragma unroll
        for (int r = 0; r < 8; ++r) {
            int gr = row0 + (wv << 4) + r + hsel;
            if (gr < rows) {
                float v = acc[r] + bn;
                if (mode >= 1) v = silu_f(v);
                size_t o = (size_t)gr * HD + ncol;
                if (mode == 2)      v *= extra[o];
                else if (mode == 3) v += extra[o];
                out[o] = v;
            }
        }
    }
}

// ---------------------------------------------------------------------------
// rbf_p[E,128] = rbf[E,6] @ W_rbf[6,128]
// ---------------------------------------------------------------------------
__global__ void k_rbf(const float* __restrict__ rbf, const float* __restrict__ Wrbf,
                      float* __restrict__ rbfp, int E)
{
    int idx = blockIdx.x * 256 + threadIdx.x;
    if (idx >= E * HD) return;
    int e = idx >> 7, n = idx & 127;
    float acc = 0.f;
    #pragma unroll
    for (int r = 0; r < 6; ++r) acc += rbf[e * 6 + r] * Wrbf[r * HD + n];
    rbfp[idx] = acc;
}

// ---------------------------------------------------------------------------
// sbf_p[T,8] = sbf[T,42] @ W_sbf[42,8]; sel[t] = angle bin; histogram counts
// ---------------------------------------------------------------------------
__global__ __launch_bounds__(256) void k_sbf(
    const float* __restrict__ sbf, const float* __restrict__ Wsbf,
    const float* __restrict__ angle, float* __restrict__ sbfp,
    int* __restrict__ sel, int* __restrict__ counts, int T)
{
    __shared__ float ws[42 * NBIL];
    for (int i = threadIdx.x; i < 42 * NBIL; i += 256) ws[i] = Wsbf[i];
    __syncthreads();
    int t = blockIdx.x * 256 + threadIdx.x;
    if (t >= T) return;
    float acc[NBIL];
    #pragma unroll
    for (int j = 0; j < NBIL; ++j) acc[j] = 0.f;
    const float* srow = sbf + (size_t)t * 42;
    for (int r = 0; r < 42; ++r) {
        float sv = srow[r];
        #pragma unroll
        for (int j = 0; j < NBIL; ++j) acc[j] += sv * ws[r * NBIL + j];
    }
    #pragma unroll
    for (int j = 0; j < NBIL; ++j) sbfp[(size_t)t * NBIL + j] = acc[j];
    int s = (int)floorf((angle[t] / 3.141593f) * 4.0f);
    s = min(max(s, 0), 3);
    sel[t] = s;
    atomicAdd(&counts[s], 1);
}

// padded (multiple-of-16) bin start offsets + scatter cursors
__global__ void k_scan(const int* __restrict__ counts, int* __restrict__ pbase,
                       int* __restrict__ cursors)
{
    if (threadIdx.x == 0 && blockIdx.x == 0) {
        int off = 0;
        for (int s = 0; s < 4; ++s) {
            pbase[s] = off; cursors[s] = off;
            off += ((counts[s] + 15) >> 4) << 4;
        }
        pbase[4] = off;
    }
}

__global__ void k_sort(const int* __restrict__ sel, int* __restrict__ cursors,
                       int* __restrict__ sorted, int T)
{
    int t = blockIdx.x * 256 + threadIdx.x;
    if (t >= T) return;
    int pos = atomicAdd(&cursors[sel[t]], 1);
    sorted[pos] = t;
}

// ---------------------------------------------------------------------------
// Cmat[s][j][m][i] = sum_l W_sel[s][m][l] * W_bil[i][j][l]   (32 matrices)
// grid = 32 matrices * 8 m-tiles; wave = n-tile
// ---------------------------------------------------------------------------
__global__ __launch_bounds__(256) void k_combine(
    const float* __restrict__ Wsel, const float* __restrict__ Wbil,
    float* __restrict__ Cmat)
{
    const int tid  = threadIdx.x;
    const int lane = tid & 31;
    const int wv   = tid >> 5;
    const int sj   = blockIdx.x >> 3;           // 0..31
    const int s    = sj >> 3, j = sj & 7;
    const int m0   = (blockIdx.x & 7) << 4;
    const int koff = (lane < 16) ? 0 : 2;
    const int hsel = (lane < 16) ? 0 : 8;
    const int mrow = m0 + (lane & 15);
    const int ncol = (wv << 4) + (lane & 15);
    const float* A = Wsel + (size_t)s * HD * HD;
    const float* Brow = Wbil + (size_t)ncol * NBIL * HD + (size_t)j * HD;
    v8f acc = {};
    #pragma unroll
    for (int k = 0; k < HD; k += 4) {
        v2f a; a.x = A[mrow * HD + k + koff]; a.y = A[mrow * HD + k + koff + 1];
        v2f b; b.x = Brow[k + koff];          b.y = Brow[k + koff + 1];
        acc = wmma_f32(a, b, acc);
    }
    float* out = Cmat + ((size_t)sj << 14);
    #pragma unroll
    for (int r = 0; r < 8; ++r)
        out[(m0 + r + hsel) * HD + ncol] = acc[r];
}

// ---------------------------------------------------------------------------
// Fused triplet kernel: gather x_kj[idx_kj] -> 8 WMMA GEMMs against Cmat[s][j]
// -> per-row sbf_p scale -> atomic scatter to agg[idx_ji].
// Block = 16 triplets x 128 cols (8 waves, one 16x16 tile each).
// ---------------------------------------------------------------------------
__global__ __launch_bounds__(256) void k_triplet(
    const float* __restrict__ xkj, const float* __restrict__ sbfp,
    const int* __restrict__ sorted, const int* __restrict__ idx_kj,
    const int* __restrict__ idx_ji, const int* __restrict__ pbase,
    const float* __restrict__ Cmat, float* __restrict__ agg)
{
    __shared__ float As[16 * LDA];
    __shared__ float sS[16 * NBIL];
    __shared__ int   sE[16];
    const int tid  = threadIdx.x;
    const int lane = tid & 31;
    const int wv   = tid >> 5;
    const int base = blockIdx.x << 4;
    if (base >= pbase[4]) return;
    int s = 0;
    if (base >= pbase[1]) s = 1;
    if (base >= pbase[2]) s = 2;
    if (base >= pbase[3]) s = 3;

    { // stage gathered rows, sbf weights, destination indices
        int r = tid >> 4;            // 0..15
        int c = (tid & 15) << 3;     // 0..120 step 8
        int t = sorted[base + r];
        if ((tid & 15) == 0) sE[r] = (t >= 0) ? idx_ji[t] : -1;
        float4 v0 = make_float4(0.f,0.f,0.f,0.f), v1 = v0;
        if (t >= 0) {
            const float* src = xkj + (size_t)idx_kj[t] * HD + c;
            v0 = *(const float4*)(src);
            v1 = *(const float4*)(src + 4);
        }
        float* p = &As[r * LDA + c];
        p[0]=v0.x; p[1]=v0.y; p[2]=v0.z; p[3]=v0.w;
        p[4]=v1.x; p[5]=v1.y; p[6]=v1.z; p[7]=v1.w;
        int j = tid & 15;
        if (j < NBIL) sS[r * NBIL + j] = (t >= 0) ? sbfp[(size_t)t * NBIL + j] : 0.f;
    }
    __syncthreads();

    const int koff = (lane < 16) ? 0 : 2;
    const int hsel = (lane < 16) ? 0 : 8;
    const int ncol = (wv << 4) + (lane & 15);
    const float* arow = &As[(lane & 15) * LDA];

    float accf[8];
    #pragma unroll
    for (int r = 0; r < 8; ++r) accf[r] = 0.f;

    for (int j = 0; j < NBIL; ++j) {
        const float* B = Cmat + ((size_t)(s * NBIL + j) << 14);
        v8f acc = {};
        #pragma unroll
        for (int k = 0; k < HD; k += 4) {
            v2f a = *(const v2f*)(arow + k + koff);
            v2f b;
            b.x = B[(k + koff) * HD + ncol];
            b.y = B[(k + koff + 1) * HD + ncol];
            acc = wmma_f32(a, b, acc);
        }
        #pragma unroll
        for (int r = 0; r < 8; ++r)
            accf[r] += acc[r] * sS[(r + hsel) * NBIL + j];
    }
    #pragma unroll
    for (int r = 0; r < 8; ++r) {
        int e = sE[r + hsel];
        if (e >= 0) atomicAdd(&agg[(size_t)e * HD + ncol], accf[r]);
    }
}

__global__ void k_add(const float* __restrict__ a, const float* __restrict__ b,
                      float* __restrict__ o, int n)
{
    int i = blockIdx.x * 256 + threadIdx.x;
    if (i < n) o[i] = a[i] + b[i];
}

// ---------------------------------------------------------------------------
extern "C" void kernel_launch(void* const* d_in, const int* in_sizes, int n_in,
                              void* d_out, int out_size, void* d_ws, size_t ws_size,
                              hipStream_t stream) {
    const float* x      = (const float*)d_in[0];
    const float* rbf    = (const float*)d_in[1];
    const float* sbf    = (const float*)d_in[2];
    const float* angle  = (const float*)d_in[3];
    const int*   idx_kj = (const int*)  d_in[4];
    const int*   idx_ji = (const int*)  d_in[5];
    const float* W_rbf  = (const float*)d_in[6];
    const float* W_sbf  = (const float*)d_in[7];
    const float* W_kj   = (const float*)d_in[8];
    const float* b_kj   = (const float*)d_in[9];
    const float* W_ji   = (const float*)d_in[10];
    const float* b_ji   = (const float*)d_in[11];
    const float* W_sel  = (const float*)d_in[12];
    const float* W_bil  = (const float*)d_in[13];
    const float* W_lin  = (const float*)d_in[14];
    const float* b_lin  = (const float*)d_in[15];
    const float* pre_W1 = (const float*)d_in[16];
    const float* pre_b1 = (const float*)d_in[17];
    const float* pre_W2 = (const float*)d_in[18];
    const float* pre_b2 = (const float*)d_in[19];
    const float* post_W1= (const float*)d_in[20];
    const float* post_b1= (const float*)d_in[21];
    const float* post_W2= (const float*)d_in[22];
    const float* post_b2= (const float*)d_in[23];

    const int E = in_sizes[0] / HD;
    const int T = in_sizes[3];
    float* hout = (float*)d_out;

    // ---- workspace carving ----
    char* ws = (char*)d_ws;
    auto aup = [](size_t v) { return (v + 255) & ~(size_t)255; };
    const size_t szEH = aup((size_t)E * HD * sizeof(float));
    size_t off = 0;
    float* B0    = (float*)(ws + off); off += szEH;   // rbf_p, later h
    float* B1    = (float*)(ws + off); off += szEH;   // x_ji
    float* B2    = (float*)(ws + off); off += szEH;   // x_kj, later tmp
    float* B3    = (float*)(ws + off); off += szEH;   // agg
    float* sbfp  = (float*)(ws + off); off += aup((size_t)T * NBIL * sizeof(float));
    int*   sel   = (int*)  (ws + off); off += aup((size_t)T * sizeof(int));
    int*   sorted= (int*)  (ws + off); off += aup((size_t)(T + 64) * sizeof(int));
    float* Cmat  = (float*)(ws + off); off += aup((size_t)32 * HD * HD * sizeof(float));
    int*   counts= (int*)  (ws + off);                 // [4]
    int*   pbase = counts + 4;                          // [5]
    int*   cursor= pbase + 5;                           // [4]

    // ---- zeroing (graph-capturable async memsets) ----
    hipMemsetAsync(counts, 0, 16 * sizeof(int), stream);
    hipMemsetAsync(sorted, 0xFF, (size_t)(T + 64) * sizeof(int), stream); // -1 sentinels
    hipMemsetAsync(B3, 0, (size_t)E * HD * sizeof(float), stream);        // agg = 0

    const int tpb = 256;
    const int gEH = (E * HD + tpb - 1) / tpb;
    const int gT  = (T + tpb - 1) / tpb;
    const int gE128 = (E + 127) / 128;

    // edge-level precompute
    k_rbf<<<gEH, tpb, 0, stream>>>(rbf, W_rbf, B0, E);
    gemm_edge<<<gE128, tpb, 0, stream>>>(x, nullptr, W_ji, b_ji, nullptr, B1, E, 1); // x_ji
    gemm_edge<<<gE128, tpb, 0, stream>>>(x, nullptr, W_kj, b_kj, B0, B2, E, 2);      // x_kj

    // triplet-level precompute
    k_sbf<<<gT, tpb, 0, stream>>>(sbf, W_sbf, angle, sbfp, sel, counts, T);
    k_scan<<<1, 32, 0, stream>>>(counts, pbase, cursor);
    k_sort<<<gT, tpb, 0, stream>>>(sel, cursor, sorted, T);
    k_combine<<<256, tpb, 0, stream>>>(W_sel, W_bil, Cmat);

    // fused gather + expert/bilinear WMMA + scatter-add
    const int gTiles = (T + 15) / 16 + 4;  // padded bins add at most 4 tiles
    k_triplet<<<gTiles, tpb, 0, stream>>>(B2, sbfp, sorted, idx_kj, idx_ji,
                                          pbase, Cmat, B3);

    // h = x_ji + agg   (B0 free now -> becomes h, B2 free -> becomes tmp)
    k_add<<<gEH, tpb, 0, stream>>>(B1, B3, B0, E * HD);

    // pre residual block
    gemm_edge<<<gE128, tpb, 0, stream>>>(B0, nullptr, pre_W1, pre_b1, nullptr, B2, E, 1);
    gemm_edge<<<gE128, tpb, 0, stream>>>(B2, nullptr, pre_W2, pre_b2, B0, B0, E, 3);

    // h = silu(h @ W_lin + b) + x   -> write to d_out
    gemm_edge<<<gE128, tpb, 0, stream>>>(B0, nullptr, W_lin, b_lin, x, hout, E, 3);

    // post residual blocks (2)
    for (int i = 0; i < 2; ++i) {
        const float* W1 = post_W1 + (size_t)i * HD * HD;
        const float* b1 = post_b1 + (size_t)i * HD;
        const float* W2 = post_W2 + (size_t)i * HD * HD;
        const float* b2 = post_b2 + (size_t)i * HD;
        gemm_edge<<<gE128, tpb, 0, stream>>>(hout, nullptr, W1, b1, nullptr, B2, E, 1);
        gemm_edge<<<gE128, tpb, 0, stream>>>(B2, nullptr, W2, b2, hout, hout, E, 3);
    }
}